// SindyNetCompEnsemble_69518340653350
// MI455X (gfx1250) — compile-verified
//
#include <hip/hip_runtime.h>
#include <math.h>

typedef __attribute__((ext_vector_type(2))) float v2f;
typedef __attribute__((ext_vector_type(8))) float v8f;

#define NROWS 4096
#define NE    64
#define INITC 192

__device__ __forceinline__ float sigmoidf(float v) {
  return 1.0f / (1.0f + __expf(-v));
}

// D = A(16x4,f32) x B(4x16,f32) + C(16x16,f32)
__device__ __forceinline__ v8f wmma4(v2f a, v2f b, v8f c) {
  return __builtin_amdgcn_wmma_f32_16x16x4_f32(false, a, false, b, (short)0, c,
                                               false, false);
}

// ---------------------------------------------------------------------------
// Pre-pass: fold the 3x3 rotations into the adjacent 3-wide linear layers.
//   encW3r[e][o][k] = sum_i encRot[e,o,i] * encW3[e,i,k]   (3x32 per expert)
//   encB3r[e][o]    = sum_i encRot[e,o,i] * encB3[e,i]
//   decW1r[e][j][i] = sum_t decW1[e,j,t]  * decRot[e,t,i]  (32x3 per expert)
// ---------------------------------------------------------------------------
__global__ void fuse_kernel(const float* __restrict__ encW3,
                            const float* __restrict__ encB3,
                            const float* __restrict__ encRot,
                            const float* __restrict__ decW1,
                            const float* __restrict__ decRot,
                            float* __restrict__ W3r, float* __restrict__ b3r,
                            float* __restrict__ W1r) {
  int tid = blockIdx.x * blockDim.x + threadIdx.x;
  int stride = gridDim.x * blockDim.x;
  for (int idx = tid; idx < 64 * 3 * 32; idx += stride) {
    int e = idx / 96, rem = idx % 96, o = rem / 32, k = rem % 32;
    float s = 0.f;
    for (int i = 0; i < 3; ++i)
      s += encRot[e * 9 + o * 3 + i] * encW3[e * 96 + i * 32 + k];
    W3r[idx] = s;
  }
  for (int idx = tid; idx < 64 * 3; idx += stride) {
    int e = idx / 3, o = idx % 3;
    float s = 0.f;
    for (int i = 0; i < 3; ++i)
      s += encRot[e * 9 + o * 3 + i] * encB3[e * 3 + i];
    b3r[idx] = s;
  }
  for (int idx = tid; idx < 64 * 32 * 3; idx += stride) {
    int e = idx / 96, rem = idx % 96, j = rem / 3, i = rem % 3;
    float s = 0.f;
    for (int t = 0; t < 3; ++t)
      s += decW1[e * 96 + j * 3 + t] * decRot[e * 9 + t * 3 + i];
    W1r[idx] = s;
  }
}

// ---------------------------------------------------------------------------
// x_stack = tile(x, 64) : (4096, 8192), vectorized as float4.
// ---------------------------------------------------------------------------
__global__ void stack_kernel(const float* __restrict__ x,
                             float* __restrict__ out) {
  const float4* x4 = (const float4*)x;
  float4* o4 = (float4*)out;
  size_t total = (size_t)NROWS * 2048;  // 8192/4 per row
  for (size_t idx = blockIdx.x * (size_t)blockDim.x + threadIdx.x; idx < total;
       idx += (size_t)gridDim.x * blockDim.x) {
    size_t b = idx >> 11;
    int c = (int)(idx & 2047);
    o4[idx] = x4[b * 32 + (c & 31)];
  }
}

// ---------------------------------------------------------------------------
// Encoder: per (expert, 16-row tile):
//   h1 = sig(X @ W1^T + b1)    16x128 @ 128x64
//   h2 = sig(h1 @ W2^T + b2)   16x64  @ 64x32
//   xe = h2 @ W3r^T + b3r      16x32  @ 32x3 (N padded to 16)
// One wave per tile; 8 waves (8 tiles) per block, all same expert.
// ---------------------------------------------------------------------------
__global__ __launch_bounds__(256) void enc_kernel(
    const float* __restrict__ x, const float* __restrict__ W1,
    const float* __restrict__ b1, const float* __restrict__ W2,
    const float* __restrict__ b2, const float* __restrict__ W3r,
    const float* __restrict__ b3r, float* __restrict__ xenc) {
  __shared__ float H1[8][16][64];
  __shared__ float H2[8][16][32];
  const int e = blockIdx.y;
  const int wave = threadIdx.x >> 5;
  const int lane = threadIdx.x & 31;
  const int row = lane & 15;  // A: M index / B,D: N index
  const int kh = lane >> 4;   // K half-select
  const int r0 = (blockIdx.x * 8 + wave) * 16;
  float(*h1)[64] = H1[wave];
  float(*h2)[32] = H2[wave];

  const float* xr = x + (size_t)(r0 + row) * 128;
  const float* W1e = W1 + e * 64 * 128;
  for (int nt = 0; nt < 4; ++nt) {
    const float* wr = W1e + (nt * 16 + row) * 128;
    v8f acc = {};
    for (int k = 0; k < 128; k += 4) {
      v2f a = {xr[k + 2 * kh], xr[k + 2 * kh + 1]};
      v2f bb = {wr[k + 2 * kh], wr[k + 2 * kh + 1]};
      acc = wmma4(a, bb, acc);
    }
    float bias = b1[e * 64 + nt * 16 + row];
#pragma unroll
    for (int v = 0; v < 8; ++v)
      h1[v + (kh ? 8 : 0)][nt * 16 + row] = sigmoidf(acc[v] + bias);
  }
  __syncthreads();

  const float* W2e = W2 + e * 32 * 64;
  for (int nt = 0; nt < 2; ++nt) {
    const float* wr = W2e + (nt * 16 + row) * 64;
    v8f acc = {};
    for (int k = 0; k < 64; k += 4) {
      v2f a = {h1[row][k + 2 * kh], h1[row][k + 2 * kh + 1]};
      v2f bb = {wr[k + 2 * kh], wr[k + 2 * kh + 1]};
      acc = wmma4(a, bb, acc);
    }
    float bias = b2[e * 32 + nt * 16 + row];
#pragma unroll
    for (int v = 0; v < 8; ++v)
      h2[v + (kh ? 8 : 0)][nt * 16 + row] = sigmoidf(acc[v] + bias);
  }
  __syncthreads();

  {
    const int n = row;
    const int nn = (n < 3) ? n : 0;  // clamp address, zero value
    const float* wr = W3r + e * 96 + nn * 32;
    v8f acc = {};
    for (int k = 0; k < 32; k += 4) {
      v2f a = {h2[row][k + 2 * kh], h2[row][k + 2 * kh + 1]};
      float w0 = wr[k + 2 * kh], w1 = wr[k + 2 * kh + 1];
      v2f bb = {(n < 3) ? w0 : 0.0f, (n < 3) ? w1 : 0.0f};
      acc = wmma4(a, bb, acc);
    }
    if (n < 3) {
      float bias = b3r[e * 3 + n];
#pragma unroll
      for (int v = 0; v < 8; ++v)
        xenc[(size_t)(r0 + v + (kh ? 8 : 0)) * INITC + e * 3 + n] =
            acc[v] + bias;
    }
  }
}

// ---------------------------------------------------------------------------
// Decoder: per (expert, 16-row tile):
//   h1 = sig(Z[:,e*3:e*3+3] @ W1r^T + b1)  16x3(padK4) @ 3x32
//   h2 = sig(h1 @ W2^T + b2)               16x32 @ 32x64
//   y  = h2 @ W3^T + b3                    16x64 @ 64x128 -> out[:, e*128 ..]
// ---------------------------------------------------------------------------
__global__ __launch_bounds__(256) void dec_kernel(
    const float* __restrict__ z, const float* __restrict__ W1r,
    const float* __restrict__ b1, const float* __restrict__ W2,
    const float* __restrict__ b2, const float* __restrict__ W3,
    const float* __restrict__ b3, float* __restrict__ out) {
  __shared__ float H1[8][16][32];
  __shared__ float H2[8][16][64];
  const int e = blockIdx.y;
  const int wave = threadIdx.x >> 5;
  const int lane = threadIdx.x & 31;
  const int row = lane & 15;
  const int kh = lane >> 4;
  const int r0 = (blockIdx.x * 8 + wave) * 16;
  float(*h1)[32] = H1[wave];
  float(*h2)[64] = H2[wave];

  {  // layer 1, single K-chunk (K=3 zero-padded to 4)
    const int k0 = 2 * kh;              // 0 or 2, always < 3
    const int k1 = 2 * kh + 1;          // 1 or 3
    const int k1c = (k1 < 3) ? k1 : 0;  // clamp address
    const float* zr = z + (size_t)(r0 + row) * INITC + e * 3;
    v2f a = {zr[k0], (k1 < 3) ? zr[k1c] : 0.0f};
    for (int nt = 0; nt < 2; ++nt) {
      const float* wr = W1r + e * 96 + (nt * 16 + row) * 3;
      v2f bb = {wr[k0], (k1 < 3) ? wr[k1c] : 0.0f};
      v8f acc = {};
      acc = wmma4(a, bb, acc);
      float bias = b1[e * 32 + nt * 16 + row];
#pragma unroll
      for (int v = 0; v < 8; ++v)
        h1[v + (kh ? 8 : 0)][nt * 16 + row] = sigmoidf(acc[v] + bias);
    }
  }
  __syncthreads();

  const float* W2e = W2 + e * 64 * 32;
  for (int nt = 0; nt < 4; ++nt) {
    const float* wr = W2e + (nt * 16 + row) * 32;
    v8f acc = {};
    for (int k = 0; k < 32; k += 4) {
      v2f a = {h1[row][k + 2 * kh], h1[row][k + 2 * kh + 1]};
      v2f bb = {wr[k + 2 * kh], wr[k + 2 * kh + 1]};
      acc = wmma4(a, bb, acc);
    }
    float bias = b2[e * 64 + nt * 16 + row];
#pragma unroll
    for (int v = 0; v < 8; ++v)
      h2[v + (kh ? 8 : 0)][nt * 16 + row] = sigmoidf(acc[v] + bias);
  }
  __syncthreads();

  const float* W3e = W3 + e * 128 * 64;
  for (int nt = 0; nt < 8; ++nt) {
    const float* wr = W3e + (nt * 16 + row) * 64;
    v8f acc = {};
    for (int k = 0; k < 64; k += 4) {
      v2f a = {h2[row][k + 2 * kh], h2[row][k + 2 * kh + 1]};
      v2f bb = {wr[k + 2 * kh], wr[k + 2 * kh + 1]};
      acc = wmma4(a, bb, acc);
    }
    float bias = b3[e * 128 + nt * 16 + row];
#pragma unroll
    for (int v = 0; v < 8; ++v)
      out[(size_t)(r0 + v + (kh ? 8 : 0)) * 8192 + e * 128 + nt * 16 + row] =
          acc[v] + bias;
  }
}

// ---------------------------------------------------------------------------
// Residual chain (comp / dcmp): z = [act(z@W^T+b), z[:, :init]] per layer,
// final z @ Wf^T + bf. Widths <= 384; LDS-resident, 16 rows per block.
// ---------------------------------------------------------------------------
struct ChainArgs {
  const float* w[6];
  const float* b[6];
  const float* wf;
  const float* bf;
  const float* zin;  // (NROWS, init_dim)
  float* zout;       // (NROWS, dfin)
  int douts[6];
  int nlayers;
  int init_dim;
  int dfin;
};

__global__ __launch_bounds__(256) void chain_kernel(ChainArgs ca) {
  __shared__ float zb[2][16][400];
  const int r0 = blockIdx.x * 16;
  const int tid = threadIdx.x;
  const int id = ca.init_dim;
  int curw = 2 * id;
  for (int idx = tid; idx < 16 * curw; idx += 256) {
    int r = idx / curw, j = idx % curw;
    zb[0][r][j] = ca.zin[(size_t)(r0 + r) * id + ((j < id) ? j : j - id)];
  }
  __syncthreads();
  int cur = 0;
  for (int l = 0; l < ca.nlayers; ++l) {
    const int dout = ca.douts[l];
    const int nxt = cur ^ 1;
    const float* W = ca.w[l];
    const float* B = ca.b[l];
    for (int idx = tid; idx < 16 * dout; idx += 256) {
      int r = idx / dout, o = idx % dout;
      const float* wr = W + o * curw;
      float acc = B[o];
      for (int i = 0; i < curw; ++i) acc += zb[cur][r][i] * wr[i];
      zb[nxt][r][o] = sigmoidf(acc);
    }
    for (int idx = tid; idx < 16 * id; idx += 256) {
      int r = idx / id, j = idx % id;
      zb[nxt][r][dout + j] = zb[cur][r][j];
    }
    __syncthreads();
    cur = nxt;
    curw = dout + id;
  }
  const int dfin = ca.dfin;
  for (int idx = tid; idx < 16 * dfin; idx += 256) {
    int r = idx / dfin, o = idx % dfin;
    const float* wr = ca.wf + o * curw;
    float acc = ca.bf[o];
    for (int i = 0; i < curw; ++i) acc += zb[cur][r][i] * wr[i];
    ca.zout[(size_t)(r0 + r) * dfin + o] = acc;
  }
}

// ---------------------------------------------------------------------------
extern "C" void kernel_launch(void* const* d_in, const int* in_sizes, int n_in,
                              void* d_out, int out_size, void* d_ws,
                              size_t ws_size, hipStream_t stream) {
  (void)in_sizes;
  (void)out_size;
  (void)ws_size;
  const float *x, *encW1, *encW2, *encW3, *encB1, *encB2, *encB3;
  const float *encRot, *decRot, *decW1, *decW2, *decW3, *decB1, *decB2, *decB3;
  const float *compW[6], *compB[6], *compWf, *compBf;
  const float *dcmpW[6], *dcmpB[6], *dcmpWf, *dcmpBf;

  if (n_in >= 43) {  // tuples flattened to individual leaves
    x = (const float*)d_in[0];
    encW1 = (const float*)d_in[1];
    encW2 = (const float*)d_in[2];
    encW3 = (const float*)d_in[3];
    encB1 = (const float*)d_in[4];
    encB2 = (const float*)d_in[5];
    encB3 = (const float*)d_in[6];
    encRot = (const float*)d_in[7];
    decRot = (const float*)d_in[8];
    decW1 = (const float*)d_in[9];
    decW2 = (const float*)d_in[10];
    decW3 = (const float*)d_in[11];
    decB1 = (const float*)d_in[12];
    decB2 = (const float*)d_in[13];
    decB3 = (const float*)d_in[14];
    for (int i = 0; i < 6; ++i) {
      compW[i] = (const float*)d_in[15 + i];
      compB[i] = (const float*)d_in[21 + i];
      dcmpW[i] = (const float*)d_in[29 + i];
      dcmpB[i] = (const float*)d_in[35 + i];
    }
    compWf = (const float*)d_in[27];
    compBf = (const float*)d_in[28];
    dcmpWf = (const float*)d_in[41];
    dcmpBf = (const float*)d_in[42];
  } else {  // tuples passed as single concatenated buffers (15 inputs)
    x = (const float*)d_in[0];
    const float* ew = (const float*)d_in[1];
    encW1 = ew;
    encW2 = ew + 64 * 64 * 128;
    encW3 = encW2 + 64 * 32 * 64;
    const float* eb = (const float*)d_in[2];
    encB1 = eb;
    encB2 = eb + 64 * 64;
    encB3 = encB2 + 64 * 32;
    encRot = (const float*)d_in[3];
    decRot = (const float*)d_in[4];
    const float* dw = (const float*)d_in[5];
    decW1 = dw;
    decW2 = dw + 64 * 32 * 3;
    decW3 = decW2 + 64 * 64 * 32;
    const float* db = (const float*)d_in[6];
    decB1 = db;
    decB2 = db + 64 * 32;
    decB3 = decB2 + 64 * 64;
    static const int cwsz[6] = {192 * 384, 96 * 384, 48 * 288,
                                24 * 240,  12 * 216, 6 * 204};
    static const int cbsz[6] = {192, 96, 48, 24, 12, 6};
    static const int dwsz[6] = {6 * 6, 12 * 9, 24 * 15, 48 * 27, 96 * 51,
                                192 * 99};
    static const int dbsz[6] = {6, 12, 24, 48, 96, 192};
    const float* cw = (const float*)d_in[7];
    const float* cb = (const float*)d_in[8];
    const float* dcw = (const float*)d_in[11];
    const float* dcb = (const float*)d_in[12];
    for (int i = 0; i < 6; ++i) {
      compW[i] = cw;
      cw += cwsz[i];
      compB[i] = cb;
      cb += cbsz[i];
      dcmpW[i] = dcw;
      dcw += dwsz[i];
      dcmpB[i] = dcb;
      dcb += dbsz[i];
    }
    compWf = (const float*)d_in[9];
    compBf = (const float*)d_in[10];
    dcmpWf = (const float*)d_in[13];
    dcmpBf = (const float*)d_in[14];
  }

  float* out = (float*)d_out;
  float* x_decomp = out;                                // (4096,192)
  float* x_encode = x_decomp + (size_t)NROWS * INITC;   // (4096,192)
  float* x_decode = x_encode + (size_t)NROWS * INITC;   // (4096,8192)
  float* x_dd = x_decode + (size_t)NROWS * 8192;        // (4096,8192)
  float* x_stack = x_dd + (size_t)NROWS * 8192;         // (4096,8192)

  float* ws = (float*)d_ws;
  float* W3r = ws;             // 64*3*32
  float* b3r = W3r + 6144;     // 64*3
  float* W1r = b3r + 192;      // 64*32*3
  float* x_comp = W1r + 6144;  // 4096*3

  fuse_kernel<<<32, 256, 0, stream>>>(encW3, encB3, encRot, decW1, decRot, W3r,
                                      b3r, W1r);
  stack_kernel<<<2048, 256, 0, stream>>>(x, x_stack);
  enc_kernel<<<dim3(32, 64), 256, 0, stream>>>(x, encW1, encB1, encW2, encB2,
                                               W3r, b3r, x_encode);

  ChainArgs ca{};
  const int cd[6] = {192, 96, 48, 24, 12, 6};
  for (int i = 0; i < 6; ++i) {
    ca.w[i] = compW[i];
    ca.b[i] = compB[i];
    ca.douts[i] = cd[i];
  }
  ca.wf = compWf;
  ca.bf = compBf;
  ca.zin = x_encode;
  ca.zout = x_comp;
  ca.nlayers = 6;
  ca.init_dim = INITC;
  ca.dfin = 3;
  chain_kernel<<<256, 256, 0, stream>>>(ca);

  ChainArgs cb{};
  const int dd[6] = {6, 12, 24, 48, 96, 192};
  for (int i = 0; i < 6; ++i) {
    cb.w[i] = dcmpW[i];
    cb.b[i] = dcmpB[i];
    cb.douts[i] = dd[i];
  }
  cb.wf = dcmpWf;
  cb.bf = dcmpBf;
  cb.zin = x_comp;
  cb.zout = x_decomp;
  cb.nlayers = 6;
  cb.init_dim = 3;
  cb.dfin = INITC;
  chain_kernel<<<256, 256, 0, stream>>>(cb);

  dec_kernel<<<dim3(32, 64), 256, 0, stream>>>(x_encode, W1r, decB1, decW2,
                                               decB2, decW3, decB3, x_decode);
  dec_kernel<<<dim3(32, 64), 256, 0, stream>>>(x_decomp, W1r, decB1, decW2,
                                               decB2, decW3, decB3, x_dd);
}